// YOLOXLoss_38087769981052
// MI455X (gfx1250) — compile-verified
//
#include <hip/hip_runtime.h>
#include <math.h>

typedef __attribute__((ext_vector_type(2))) float v2f;
typedef __attribute__((ext_vector_type(8))) float v8f;

#define NUM_ANCHORS 8400
#define NTILES      525          // 8400 / 16
#define NCLS        80
#define MAXT        50
#define IMG         640.0f
#define DIST_THRESH 4096.0f      // 64^2

// Stable BCE-with-logits: max(x,0) - x*t + log1p(exp(-|x|))
// == -(t*log_sigmoid(x) + (1-t)*log_sigmoid(-x))
__device__ __forceinline__ float bce_logits(float x, float t) {
    return fmaxf(x, 0.0f) - x * t + log1pf(expf(-fabsf(x)));
}

__device__ __forceinline__ void anchor_xy(int n, float& ax, float& ay) {
    int idx, dim, s;
    if (n < 6400)      { idx = n;        dim = 80; s = 8;  }
    else if (n < 8000) { idx = n - 6400; dim = 40; s = 16; }
    else               { idx = n - 8000; dim = 20; s = 32; }
    ax = (float)(idx % dim) * (float)s + (float)s * 0.5f;
    ay = (float)(idx / dim) * (float)s + (float)s * 0.5f;
}

__global__ void yolox_zero_kernel(float* accum) {
    accum[threadIdx.x] = 0.0f;   // 64 images * 4 accumulators = 256 floats
}

// One wave handles a 16-anchor column tile of one image.
// WMMA computes dot(c_m, a_n) for 64 (padded) targets x 16 anchors; then
// d2 = |c|^2 + |a|^2 - 2*dot, min/argmin over m, then fused loss evaluation.
__global__ __launch_bounds__(256) void yolox_match_loss_kernel(
    const float* __restrict__ pred,   // [64, 8400, 85]
    const float* __restrict__ tgt,    // [64, 50, 5]
    float* __restrict__ accum)        // [64, 4] = {obj, cls, box, npos}
{
    const int b   = blockIdx.y;
    const int tid = threadIdx.x;

    __shared__ float s_cx[64], s_cy[64], s_cn[64], s_cls[64], s_w[64], s_h[64];
    if (tid < 64) {
        float cls = -1.0f, cx = 0.0f, cy = 0.0f, w = 0.0f, h = 0.0f;
        float cn  = __builtin_inff();                 // padded / invalid -> d2 = inf
        if (tid < MAXT) {
            const float* t = tgt + ((size_t)b * MAXT + tid) * 5;
            cls = t[0];
            cx = t[1] * IMG; cy = t[2] * IMG; w = t[3] * IMG; h = t[4] * IMG;
            if (cls >= 0.0f) cn = cx * cx + cy * cy;  // valid target
        }
        s_cx[tid] = cx; s_cy[tid] = cy; s_cn[tid] = cn;
        s_cls[tid] = cls; s_w[tid] = w; s_h[tid] = h;
    }
    __syncthreads();

    const int wave  = tid >> 5;
    const int lane  = tid & 31;
    const int ntile = blockIdx.x * 8 + wave;
    if (ntile >= NTILES) return;                      // whole-wave uniform exit

    const int nLoc = lane & 15;
    const int hi   = lane >> 4;                       // 0: K=0..1 / M=v ; 1: K=2..3 / M=v+8
    const int n    = ntile * 16 + nLoc;

    float ax, ay; anchor_xy(n, ax, ay);
    const float anorm = ax * ax + ay * ay;

    // B matrix 4x16: lanes 0-15 hold rows K=0,1 (the anchor coords); K=2,3 zero.
    v2f bvec;
    bvec.x = hi ? 0.0f : ax;
    bvec.y = hi ? 0.0f : ay;
    v8f zc = {0.0f, 0.0f, 0.0f, 0.0f, 0.0f, 0.0f, 0.0f, 0.0f};

    float minv = __builtin_inff();
    int   argm = 0;
    #pragma unroll
    for (int t4 = 0; t4 < 4; ++t4) {                  // 4 target tiles of 16 (64 padded)
        const int ma = t4 * 16 + nLoc;
        v2f avec;                                     // A 16x4: lanes 0-15 rows, K=0,1
        avec.x = hi ? 0.0f : s_cx[ma];
        avec.y = hi ? 0.0f : s_cy[ma];
        v8f d = __builtin_amdgcn_wmma_f32_16x16x4_f32(
            false, avec, false, bvec, (short)0, zc, false, false);
        #pragma unroll
        for (int v = 0; v < 8; ++v) {                 // lane holds M = t4*16 + v + hi*8
            const int m = t4 * 16 + v + hi * 8;
            const float d2 = s_cn[m] + anorm - 2.0f * d[v];
            if (d2 < minv) { minv = d2; argm = m; }   // ascending m -> first-index ties
        }
    }
    // merge the two lane halves (each tracked a disjoint half of the m's)
    {
        const float om = __shfl_xor(minv, 16, 32);
        const int   oa = __shfl_xor(argm, 16, 32);
        if (om < minv || (om == minv && oa < argm)) { minv = om; argm = oa; }
    }
    const bool  pos  = minv < DIST_THRESH;
    const float tpos = pos ? 1.0f : 0.0f;

    const float* base = pred + ((size_t)b * NUM_ANCHORS + n) * (5 + NCLS);

    float obj_acc = 0.0f, cls_acc = 0.0f, box_acc = 0.0f, np_acc = 0.0f;
    if (hi == 0) {                                    // one lane per anchor for obj/box
        const float xo = base[4];
        obj_acc = bce_logits(xo, tpos);
        np_acc  = tpos;
    }
    if (pos) {
        const int gcls = (int)s_cls[argm];
        const int c0   = hi ? 40 : 0;                 // split 80 classes across halves
        #pragma unroll 8
        for (int k = 0; k < 40; ++k) {
            const int c = c0 + k;
            cls_acc += bce_logits(base[5 + c], (c == gcls) ? 1.0f : 0.0f);
        }
        if (hi == 0) {                                // IoU loss
            const float pcx = base[0], pcy = base[1];
            const float pw = expf(base[2]), ph = expf(base[3]);
            const float px1 = pcx - 0.5f * pw, px2 = pcx + 0.5f * pw;
            const float py1 = pcy - 0.5f * ph, py2 = pcy + 0.5f * ph;
            const float gcx = s_cx[argm], gcy = s_cy[argm];
            const float gw = s_w[argm],  gh = s_h[argm];
            const float tx1 = gcx - 0.5f * gw, tx2 = gcx + 0.5f * gw;
            const float ty1 = gcy - 0.5f * gh, ty2 = gcy + 0.5f * gh;
            const float iw = fminf(px2, tx2) - fmaxf(px1, tx1);
            const float ih = fminf(py2, ty2) - fmaxf(py1, ty1);
            const float inter = fmaxf(iw, 0.0f) * fmaxf(ih, 0.0f);
            const float pa = (px2 - px1) * (py2 - py1);
            const float ta = (tx2 - tx1) * (ty2 - ty1);
            box_acc = 1.0f - inter / (pa + ta - inter + 1e-6f);
        }
    }
    // full-wave butterfly reduction (all 32 lanes active here)
    #pragma unroll
    for (int off = 16; off; off >>= 1) {
        obj_acc += __shfl_xor(obj_acc, off, 32);
        cls_acc += __shfl_xor(cls_acc, off, 32);
        box_acc += __shfl_xor(box_acc, off, 32);
        np_acc  += __shfl_xor(np_acc,  off, 32);
    }
    if (lane == 0) {
        atomicAdd(&accum[b * 4 + 0], obj_acc);
        atomicAdd(&accum[b * 4 + 1], cls_acc);
        atomicAdd(&accum[b * 4 + 2], box_acc);
        atomicAdd(&accum[b * 4 + 3], np_acc);
    }
}

__global__ void yolox_finalize_kernel(const float* __restrict__ accum,
                                      float* __restrict__ out)
{
    __shared__ float sm[4][64];
    const int i = threadIdx.x;                        // 64 threads, one per image
    const float o  = accum[i * 4 + 0];
    const float c  = accum[i * 4 + 1];
    const float bx = accum[i * 4 + 2];
    const float np = accum[i * 4 + 3];
    const float obj = o * (1.0f / (float)NUM_ANCHORS);
    const float cls = (np > 0.0f) ? c / (np * (float)NCLS) : 0.0f;
    const float box = (np > 0.0f) ? bx / np : 0.0f;
    sm[0][i] = obj; sm[1][i] = cls; sm[2][i] = box; sm[3][i] = np;
    __syncthreads();
    for (int off = 32; off > 0; off >>= 1) {
        if (i < off) {
            sm[0][i] += sm[0][i + off];
            sm[1][i] += sm[1][i + off];
            sm[2][i] += sm[2][i + off];
            sm[3][i] += sm[3][i + off];
        }
        __syncthreads();
    }
    if (i == 0) {
        const float OBJ = sm[0][0], CLS = sm[1][0], BOX = sm[2][0], NP = sm[3][0];
        out[0] = 5.0f * BOX + OBJ + CLS;  // total
        out[1] = BOX;                     // box_loss
        out[2] = OBJ;                     // obj_loss
        out[3] = CLS;                     // cls_loss
        out[4] = NP;                      // num_fg
    }
}

extern "C" void kernel_launch(void* const* d_in, const int* in_sizes, int n_in,
                              void* d_out, int out_size, void* d_ws, size_t ws_size,
                              hipStream_t stream) {
    const float* pred = (const float*)d_in[0];   // [64, 8400, 85] fp32
    const float* tgt  = (const float*)d_in[1];   // [64, 50, 5] fp32
    float* accum = (float*)d_ws;                 // 256 floats of scratch
    float* out   = (float*)d_out;                // 5 floats

    yolox_zero_kernel<<<dim3(1), dim3(256), 0, stream>>>(accum);
    dim3 grid((NTILES + 7) / 8, 64);
    yolox_match_loss_kernel<<<grid, dim3(256), 0, stream>>>(pred, tgt, accum);
    yolox_finalize_kernel<<<dim3(1), dim3(64), 0, stream>>>(accum, out);
}